// SocialGCN_87866440942256
// MI455X (gfx1250) — compile-verified
//
#include <hip/hip_runtime.h>
#include <hip/hip_bf16.h>

typedef float v2f __attribute__((ext_vector_type(2)));
typedef float v8f __attribute__((ext_vector_type(8)));

#define EMBED 128
#define MT_PER_WAVE 4   // M-tiles swept per wave against register-resident B

// -------------------------------------------------------------------------
// Layer-0 passthrough: out0 = user_embeds  (float4 grid-stride copy)
// -------------------------------------------------------------------------
__global__ __launch_bounds__(256) void copy_f4_kernel(const float4* __restrict__ src,
                                                      float4* __restrict__ dst,
                                                      int n4) {
    int i = blockIdx.x * blockDim.x + threadIdx.x;
    int stride = gridDim.x * blockDim.x;
    for (; i < n4; i += stride) dst[i] = src[i];
}

// -------------------------------------------------------------------------
// Dense GEMM  Y[M,128] = X[M,128] @ W[128,128]  via V_WMMA_F32_16X16X4_F32.
//
// Each wave32 owns one 16-column slice of W (n-tile) and pins the FULL
// K=128 B-operand for that slice in 64 VGPRs (v2f b[32]).  It then sweeps
// MT_PER_WAVE consecutive 16-row M-tiles, doing 32 WMMAs per tile with only
// one b64 A-load per WMMA.  B is read from L2 once per wave instead of once
// per tile (4x less B traffic, ~1 load per WMMA instead of 3).
//
// A (16x4 f32) layout (ISA 7.12.2): lanes 0-15 -> row M=lane, K={k,k+1};
// lanes 16-31 -> row M=lane-16, K={k+2,k+3}.  B mirrored per column.
// C/D: VGPR j = row m0 + j (+8 for lanes 16-31), col = lane&15.
// Tile-count loop bound is wave-uniform -> EXEC stays all-1s (WMMA req).
// -------------------------------------------------------------------------
__global__ __launch_bounds__(128) void gemm_wmma_f32_kernel(const float* __restrict__ X,
                                                            const float* __restrict__ W,
                                                            float* __restrict__ Y,
                                                            int mTiles) {
    const int lane = threadIdx.x & 31;
    const int wave = threadIdx.x >> 5;                 // 0..3
    const int half = lane >> 4;                        // 0/1: K sub-pair select
    const int l16  = lane & 15;

    const int nTile = ((blockIdx.x & 1) << 2) + wave;  // 0..7 -> N in [0,128)
    const int mBase = (blockIdx.x >> 1) * MT_PER_WAVE; // first M-tile of sweep
    if (mBase >= mTiles) return;
    const int nt = min(MT_PER_WAVE, mTiles - mBase);   // wave-uniform

    const int n0 = nTile * 16;

    // ---- Pin B (K=128 x N=16 slice of W) in registers: 64 VGPRs ----
    const float* __restrict__ wcol = W + (size_t)(half << 1) * EMBED + n0 + l16;
    v2f b[32];
    #pragma unroll
    for (int kq = 0; kq < 32; ++kq) {
        b[kq].x = wcol[(size_t)(kq * 4 + 0) * EMBED];
        b[kq].y = wcol[(size_t)(kq * 4 + 1) * EMBED];
    }

    // ---- Sweep M-tiles against register-resident B ----
    for (int t = 0; t < nt; ++t) {
        const int m0 = (mBase + t) * 16;
        const float* __restrict__ xrow = X + (size_t)(m0 + l16) * EMBED + (half << 1);

        v8f c = {};
        #pragma unroll
        for (int kq = 0; kq < 32; ++kq) {
            v2f a;
            a.x = xrow[kq * 4 + 0];                    // contiguous pair -> b64 load
            a.y = xrow[kq * 4 + 1];
            c = __builtin_amdgcn_wmma_f32_16x16x4_f32(
                    /*neg_a=*/false, a, /*neg_b=*/false, b[kq],
                    /*c_mod=*/(short)0, c, /*reuse_a=*/false, /*reuse_b=*/false);
        }

        float* __restrict__ yrow = Y + (size_t)(m0 + (half << 3)) * EMBED + n0 + l16;
        #pragma unroll
        for (int j = 0; j < 8; ++j) yrow[(size_t)j * EMBED] = c[j];
    }
}

// Scalar tail for M % 16 rows (not launched when M divisible by 16).
__global__ __launch_bounds__(128) void gemm_tail_kernel(const float* __restrict__ X,
                                                        const float* __restrict__ W,
                                                        float* __restrict__ Y,
                                                        int mStart, int M) {
    int r = mStart + blockIdx.x;
    if (r >= M) return;
    int n = threadIdx.x;                               // 0..127
    float acc = 0.f;
    const float* xrow = X + (size_t)r * EMBED;
    for (int k = 0; k < EMBED; ++k) acc += xrow[k] * W[(size_t)k * EMBED + n];
    Y[(size_t)r * EMBED + n] = acc;
}

// -------------------------------------------------------------------------
// SpMM + residual:  out[r] = x[r] + sum_{e in row r} vals[e] * xt[cols[e]]
// s_rows is sorted -> binary-search row boundaries. One 128-thread block per
// row; thread t owns embedding dim t. Gathers of xt hit the 192MB L2.
// -------------------------------------------------------------------------
__global__ __launch_bounds__(128) void spmm_residual_kernel(const float* __restrict__ X,
                                                            const float* __restrict__ XT,
                                                            const int* __restrict__ rows,
                                                            const int* __restrict__ cols,
                                                            const float* __restrict__ vals,
                                                            float* __restrict__ Y,
                                                            int nEdges) {
    __shared__ int se[2];
    const int r = blockIdx.x;
    const int t = threadIdx.x;

    if (t == 0) {
        // lower_bound(rows, r)
        int lo = 0, hi = nEdges;
        while (lo < hi) { int mid = (lo + hi) >> 1; if (rows[mid] < r) lo = mid + 1; else hi = mid; }
        se[0] = lo;
        // lower_bound(rows, r+1), starting at lo
        int lo2 = lo, hi2 = nEdges;
        while (lo2 < hi2) { int mid = (lo2 + hi2) >> 1; if (rows[mid] <= r) lo2 = mid + 1; else hi2 = mid; }
        se[1] = lo2;
    }
    __syncthreads();

    const int start = se[0], end = se[1];
    float acc = 0.f;
    for (int e = start; e < end; ++e) {
        const int   c = cols[e];   // uniform across block
        const float v = vals[e];   // uniform across block
        acc += v * XT[(size_t)c * EMBED + t];
    }
    const size_t idx = (size_t)r * EMBED + t;
    Y[idx] = X[idx] + acc;
}

// -------------------------------------------------------------------------
// Launch: out = [x0, x1, x2] with x_{l+1} = x_l + S @ (x_l @ W_l)
// -------------------------------------------------------------------------
extern "C" void kernel_launch(void* const* d_in, const int* in_sizes, int n_in,
                              void* d_out, int out_size, void* d_ws, size_t ws_size,
                              hipStream_t stream) {
    const float* X0   = (const float*)d_in[0];   // [M,128]
    const int*   rows = (const int*)  d_in[1];   // [E] sorted
    const int*   cols = (const int*)  d_in[2];   // [E]
    const float* vals = (const float*)d_in[3];   // [E]
    const float* W0   = (const float*)d_in[4];   // [128,128]
    const float* W1   = (const float*)d_in[5];   // [128,128]

    const int M = in_sizes[0] / EMBED;           // 100000
    const int E = in_sizes[1];                   // 1.6M

    float* out  = (float*)d_out;
    float* out0 = out;
    float* out1 = out + (size_t)M * EMBED;
    float* out2 = out + (size_t)2 * M * EMBED;
    float* xt   = (float*)d_ws;                  // [M,128] scratch (51.2 MB)

    // Layer 0: identity copy
    {
        const int n4 = (M * EMBED) / 4;
        const int blocks = (n4 + 255) / 256;
        copy_f4_kernel<<<blocks, 256, 0, stream>>>((const float4*)X0, (float4*)out0, n4);
    }

    const int mTiles  = M / 16;
    const int mTail   = M - mTiles * 16;
    const int mSupers = (mTiles + MT_PER_WAVE - 1) / MT_PER_WAVE;
    const int gemmBlocks = mSupers * 2;          // x2: two n-tile halves per block row

    // ---- Layer 1 ----
    gemm_wmma_f32_kernel<<<gemmBlocks, 128, 0, stream>>>(X0, W0, xt, mTiles);
    if (mTail > 0)
        gemm_tail_kernel<<<mTail, 128, 0, stream>>>(X0, W0, xt, mTiles * 16, M);
    spmm_residual_kernel<<<M, 128, 0, stream>>>(X0, xt, rows, cols, vals, out1, E);

    // ---- Layer 2 ----
    gemm_wmma_f32_kernel<<<gemmBlocks, 128, 0, stream>>>(out1, W1, xt, mTiles);
    if (mTail > 0)
        gemm_tail_kernel<<<mTail, 128, 0, stream>>>(out1, W1, xt, mTiles * 16, M);
    spmm_residual_kernel<<<M, 128, 0, stream>>>(out1, xt, rows, cols, vals, out2, E);
}